// EternalNeuralLayer_50964081934644
// MI455X (gfx1250) — compile-verified
//
#include <hip/hip_runtime.h>
#include <math.h>

typedef float v2f __attribute__((ext_vector_type(2)));
typedef float v8f __attribute__((ext_vector_type(8)));

// -------- Kernel 1: probs[i] = (prod cos(ew[d,i,k]))^2 / n  (analytic reduction of scan) --------
// The scan state stays uniform along axis 1 (circular roll of a uniform row is itself),
// so each gate just multiplies row i by cos(angle[i]).
__global__ __launch_bounds__(256) void eternal_probs_kernel(
    const float* __restrict__ ew,   // (D, N, M)
    float* __restrict__ probs,      // (M,)
    int D, int N, int M, float inv_n) {
  int i = blockIdx.x * blockDim.x + threadIdx.x;
  if (i >= M) return;
  float p = 1.0f;
  for (int d = 0; d < D; ++d) {
    const float* a = ew + (size_t)d * N * M + (size_t)i * M;
    p *= cosf(a[0]);
    p *= cosf(a[1]);
    p *= cosf(a[2]);
  }
  probs[i] = p * p * inv_n;
}

// -------- Kernel 2: out = tanh(X @ W + bias + probs) via V_WMMA_F32_16X16X4_F32 --------
// X: (Brows, K) row-major; W: (K, M) row-major; out: (Brows, M)
// Block = 256 threads = 8 wave32s arranged 2(M) x 4(N). Block tile 128 x 256.
// Wave tile 64x64 = 4x4 accumulators of 16x16 (128 accum VGPRs).
// All global accesses use 32-bit element indices off the scalar kernel-arg base
// so the compiler emits global_load (saddr + voffset + imm) instead of flat_load.
__global__ __launch_bounds__(256) void gemm_bias_tanh_kernel(
    const float* __restrict__ X,
    const float* __restrict__ W,
    const float* __restrict__ bias,
    const float* __restrict__ probs,
    float* __restrict__ out,
    int Brows, int K, int M) {
  (void)Brows;
  const int lane = threadIdx.x & 31;
  const int wave = threadIdx.x >> 5;
  const int l15  = lane & 15;
  const int lhi  = lane >> 4;  // 0 or 1

  const int m_base = blockIdx.y * 128 + (wave >> 2) * 64;  // wave rows
  const int n_base = blockIdx.x * 256 + (wave & 3) * 64;   // wave cols

  v8f acc[4][4];
#pragma unroll
  for (int t = 0; t < 4; ++t)
#pragma unroll
    for (int u = 0; u < 4; ++u)
      acc[t][u] = v8f{};

  // A-frag layout (16x4 f32): lane L -> row m0+(L&15), K = k0 + 2*(L>>4) + {0,1}
  int a_idx[4];
#pragma unroll
  for (int t = 0; t < 4; ++t) {
    a_idx[t] = (m_base + t * 16 + l15) * K + lhi * 2;
    __builtin_assume(a_idx[t] >= 0);
  }

  // B-frag layout (4x16 f32): lane L -> col n0+(L&15), K = k0 + 2*(L>>4) + {0,1}
  int b_idx[4];
#pragma unroll
  for (int u = 0; u < 4; ++u) {
    b_idx[u] = (lhi * 2) * M + n_base + u * 16 + l15;
    __builtin_assume(b_idx[u] >= 0);
  }

  int bk = 0;  // k0 * M, kept as a 32-bit running element offset
#pragma unroll 4
  for (int k0 = 0; k0 < K; k0 += 4) {
    v2f a[4];
#pragma unroll
    for (int t = 0; t < 4; ++t)
      a[t] = *(const v2f*)(X + (a_idx[t] + k0));   // global_load_b64

    v2f b[4];
#pragma unroll
    for (int u = 0; u < 4; ++u) {
      b[u].x = W[b_idx[u] + bk];                   // row k0 + 2*lhi
      b[u].y = W[b_idx[u] + bk + M];               // row k0 + 2*lhi + 1
    }
    bk += 4 * M;

#pragma unroll
    for (int t = 0; t < 4; ++t)
#pragma unroll
      for (int u = 0; u < 4; ++u)
        acc[t][u] = __builtin_amdgcn_wmma_f32_16x16x4_f32(
            /*neg_a=*/false, a[t], /*neg_b=*/false, b[u],
            /*c_mod=*/(short)0, acc[t][u],
            /*reuse_a=*/false, /*reuse_b=*/false);
  }

  // Epilogue: C/D layout -> lane L, VGPR v holds (row = v + 8*(L>>4), col = L&15)
#pragma unroll
  for (int u = 0; u < 4; ++u) {
    const int col = n_base + u * 16 + l15;
    const float bp = bias[col] + probs[col];
#pragma unroll
    for (int t = 0; t < 4; ++t) {
      const int row0 = m_base + t * 16 + lhi * 8;
#pragma unroll
      for (int v = 0; v < 8; ++v) {
        float r = tanhf(acc[t][u][v] + bp);
        out[(size_t)(row0 + v) * M + col] = r;
      }
    }
  }
}

extern "C" void kernel_launch(void* const* d_in, const int* in_sizes, int n_in,
                              void* d_out, int out_size, void* d_ws, size_t ws_size,
                              hipStream_t stream) {
  (void)n_in; (void)out_size; (void)ws_size;
  const float* x  = (const float*)d_in[0];  // (B, N)
  const float* ew = (const float*)d_in[1];  // (D, N, M)
  // d_in[2] = eternal_biases: unused by the reference
  const float* cw = (const float*)d_in[3];  // (N, M)
  const float* cb = (const float*)d_in[4];  // (M,)
  float* out   = (float*)d_out;
  float* probs = (float*)d_ws;              // M floats of scratch

  const int M = in_sizes[4];                // 2048
  const int K = in_sizes[3] / M;            // 2048 (N in reference)
  const int B = in_sizes[0] / K;            // 4096
  const int D = in_sizes[1] / (K * M);      // 9

  {
    int blocks = (M + 255) / 256;
    eternal_probs_kernel<<<blocks, 256, 0, stream>>>(ew, probs, D, K, M,
                                                     1.0f / (float)K);
  }
  {
    dim3 grid(M / 256, B / 128);  // 8 x 32 blocks (M=2048, B=4096)
    gemm_bias_tanh_kernel<<<grid, dim3(256, 1, 1), 0, stream>>>(
        x, cw, cb, probs, out, B, K, M);
  }
}